// AlphaGridMask_60060822667739
// MI455X (gfx1250) — compile-verified
//
#include <hip/hip_runtime.h>

// Trilinear grid_sample (align_corners=true, zero padding) over a 256^3 volume,
// with AABB normalize + inf-norm space contraction, matching the JAX reference.
//
// MI455X reasoning: 0.4 GFLOP vs 160 MiB streamed + 67M scattered taps ->
// bandwidth / L2-gather bound; WMMA inapplicable. Volume (64 MiB) < L2
// (192 MB): keep it L2-resident (RT gathers), mark read-once xyz / write-once
// out as non-temporal (th:TH_NT). Adjacent x-taps are fused into b64 pair
// loads: 4 gathers/point instead of 8, halving L2 request count.

typedef float v4f __attribute__((ext_vector_type(4)));
typedef float v2f __attribute__((ext_vector_type(2)));

#define VOLD 256
#define ITEMS 4

__global__ __launch_bounds__(256, 1) void alpha_grid_sample_kernel(
    const v4f* __restrict__ xyz,      // [N] (x,y,z,w)
    const float* __restrict__ vol,    // [256,256,256] z-major
    const float* __restrict__ aabb,   // [2,3]
    const int* __restrict__ contract, // [1]
    float* __restrict__ out,          // [N]
    int n)
{
    // Uniform params -> scalarized s_loads.
    const float a0x = aabb[0], a0y = aabb[1], a0z = aabb[2];
    const float a1x = aabb[3], a1y = aabb[4], a1z = aabb[5];
    const int   cs  = contract[0];
    const float gxs = 2.0f / (a1x - a0x);
    const float gys = 2.0f / (a1y - a0y);
    const float gzs = 2.0f / (a1z - a0z);

    const int tid  = threadIdx.x;
    const int base = blockIdx.x * (256 * ITEMS);

#pragma unroll
    for (int j = 0; j < ITEMS; ++j) {
        const int i = base + j * 256 + tid;
        if (i >= n) break;

        // Pull next chunk's xyz line toward the WGP (global_prefetch_b8,
        // near-cache / non-speculative).
        if (j + 1 < ITEMS) __builtin_prefetch((const void*)&xyz[i + 256], 0, 3);

        // Read-once stream: non-temporal b128 load (th:TH_NT).
        v4f p = __builtin_nontemporal_load(&xyz[i]);

        // ---- normalize into [-1,1] inside aabb ----
        float cx = (p.x - a0x) * gxs - 1.0f;
        float cy = (p.y - a0y) * gys - 1.0f;
        float cz = (p.z - a0z) * gzs - 1.0f;

        // ---- inf-norm space contraction (single division) ----
        if (cs) {
            float dist = fmaxf(fabsf(cx), fmaxf(fabsf(cy), fabsf(cz))) + 1e-8f;
            float rd   = 1.0f / dist;
            float s    = (dist > 1.0f) ? (2.0f - rd) : dist;
            float m    = s * 0.5f * rd;      // where(...)*0.5 * (coords/dist)
            cx *= m; cy *= m; cz *= m;
        }

        // ---- grid_sample index math (align_corners=true) ----
        float ix = (cx + 1.0f) * 0.5f * (float)(VOLD - 1);
        float iy = (cy + 1.0f) * 0.5f * (float)(VOLD - 1);
        float iz = (cz + 1.0f) * 0.5f * (float)(VOLD - 1);

        float x0f = floorf(ix), y0f = floorf(iy), z0f = floorf(iz);
        float tx = ix - x0f, ty = iy - y0f, tz = iz - z0f;
        int x0 = (int)x0f, y0 = (int)y0f, z0 = (int)z0f;

        // Zero-padding handled by zeroing tap WEIGHTS (w*0 == 0*v for the
        // finite volume data), so gathers can use clamped indices freely.
        bool bx0 = (unsigned)x0       < (unsigned)VOLD;
        bool bx1 = (unsigned)(x0 + 1) < (unsigned)VOLD;
        bool by0 = (unsigned)y0       < (unsigned)VOLD;
        bool by1 = (unsigned)(y0 + 1) < (unsigned)VOLD;
        bool bz0 = (unsigned)z0       < (unsigned)VOLD;
        bool bz1 = (unsigned)(z0 + 1) < (unsigned)VOLD;

        float wxl = bx0 ? (1.0f - tx) : 0.0f;
        float wxh = bx1 ? tx          : 0.0f;
        float wyl = by0 ? (1.0f - ty) : 0.0f;
        float wyh = by1 ? ty          : 0.0f;
        float wzl = bz0 ? (1.0f - tz) : 0.0f;
        float wzh = bz1 ? tz          : 0.0f;

        // Clamped row bases and paired-x base (x0 and x0+1 are adjacent!).
        int yc0 = min(max(y0, 0), VOLD - 1), yc1 = min(max(y0 + 1, 0), VOLD - 1);
        int zc0 = min(max(z0, 0), VOLD - 1), zc1 = min(max(z0 + 1, 0), VOLD - 1);
        int xb  = min(max(x0, 0), VOLD - 2);              // pair base: 0..254

        int r00 = (zc0 * VOLD + yc0) * VOLD + xb;
        int r01 = (zc0 * VOLD + yc1) * VOLD + xb;
        int r10 = (zc1 * VOLD + yc0) * VOLD + xb;
        int r11 = (zc1 * VOLD + yc1) * VOLD + xb;

        // 4 independent b64 pair gathers (L2-resident volume, RT hint).
        v2f p00, p01, p10, p11;
        __builtin_memcpy(&p00, vol + r00, sizeof(v2f));
        __builtin_memcpy(&p01, vol + r01, sizeof(v2f));
        __builtin_memcpy(&p10, vol + r10, sizeof(v2f));
        __builtin_memcpy(&p11, vol + r11, sizeof(v2f));

        // Pair -> (x0-tap, x1-tap) element selection. Only edge cases shift:
        //   x0 == 255 -> xb=254, x0-tap is pair.y (x1-tap weight already 0)
        //   x0 == -1  -> xb=0,   x1-tap is pair.x (x0-tap weight already 0)
        bool hi0 = (x0 == VOLD - 1);
        bool lo1 = (x0 == -1);
        float v000 = hi0 ? p00.y : p00.x,  v001 = lo1 ? p00.x : p00.y;
        float v010 = hi0 ? p01.y : p01.x,  v011 = lo1 ? p01.x : p01.y;
        float v100 = hi0 ? p10.y : p10.x,  v101 = lo1 ? p10.x : p10.y;
        float v110 = hi0 ? p11.y : p11.x,  v111 = lo1 ? p11.x : p11.y;

        // Same accumulation order as reference: z outer, y mid, x inner.
        float wll = wzl * wyl, wlh = wzl * wyh;
        float whl = wzh * wyl, whh = wzh * wyh;
        float acc;
        acc  = wll * wxl * v000;
        acc += wll * wxh * v001;
        acc += wlh * wxl * v010;
        acc += wlh * wxh * v011;
        acc += whl * wxl * v100;
        acc += whl * wxh * v101;
        acc += whh * wxl * v110;
        acc += whh * wxh * v111;

        // Write-once stream: non-temporal store (th:TH_NT).
        __builtin_nontemporal_store(acc, &out[i]);
    }
}

extern "C" void kernel_launch(void* const* d_in, const int* in_sizes, int n_in,
                              void* d_out, int out_size, void* d_ws, size_t ws_size,
                              hipStream_t stream) {
    const v4f*   xyz  = (const v4f*)d_in[0];    // xyz_sampled [N,4] f32
    const float* vol  = (const float*)d_in[1];  // alpha_volume [256^3] f32
    const float* aabb = (const float*)d_in[2];  // aabb [2,3] f32
    const int*   cs   = (const int*)d_in[3];    // contract_space scalar
    float*       out  = (float*)d_out;          // [N] f32

    const int n = in_sizes[0] / 4;
    const int block = 256;
    const int per_block = block * ITEMS;
    const int blocks = (n + per_block - 1) / per_block;

    alpha_grid_sample_kernel<<<blocks, block, 0, stream>>>(xyz, vol, aabb, cs, out, n);
}